// scFoundationTransform_57629871178234
// MI455X (gfx1250) — compile-verified
//
#include <hip/hip_runtime.h>
#include <hip/hip_bf16.h>
#include <math.h>

// Problem constants (match reference)
#define N_CELLS 4096
#define N_PANEL 18000
#define N_GENES 19264
#define N_OUTC  19266   // 19264 + 2 totalcount columns

typedef float v2f __attribute__((ext_vector_type(2)));
typedef float v8f __attribute__((ext_vector_type(8)));

// Async global->LDS b128 operand types (from hipcc diagnostic: param 0 is
// "__vector_size__(4*sizeof(int)) int __device__ *", i.e. non-const int4* in AS1).
typedef int vec4i __attribute__((vector_size(16)));
typedef __attribute__((address_space(1))) vec4i* gbl4_t;
typedef __attribute__((address_space(3))) vec4i* lds4_t;

// ---------------------------------------------------------------------------
// Kernel 0: zero the weight histogram
// ---------------------------------------------------------------------------
__global__ void scf_zero_w(float* __restrict__ w) {
    int i = blockIdx.x * blockDim.x + threadIdx.x;
    if (i < N_PANEL) w[i] = 0.0f;
}

// ---------------------------------------------------------------------------
// Kernel 1: w[j] = #{ g : gather_idx[g]==j && !missing[g] }
// Float atomic adds of 1.0f: all partial sums are exact integers <= 19264,
// so the result is deterministic regardless of atomic ordering.
// ---------------------------------------------------------------------------
__global__ void scf_build_w(const int* __restrict__ gidx,
                            const unsigned char* __restrict__ missing,
                            float* __restrict__ w) {
    int g = blockIdx.x * blockDim.x + threadIdx.x;
    if (g < N_GENES && missing[g] == 0) {
        atomicAdd(&w[gidx[g]], 1.0f);
    }
}

// ---------------------------------------------------------------------------
// Kernel 2: counts = X @ w  (4096x18000 GEMV) via V_WMMA_F32_16X16X4_F32.
// One wave (32 threads) per 16-row tile; 256 blocks.
// A layout (ISA 16x4 f32): lanes 0-15 -> K=0,1 ; lanes 16-31 -> K=2,3; M=lane%16.
// B is w broadcast across all 16 N columns, so every D column holds the dot.
// ---------------------------------------------------------------------------
__global__ __launch_bounds__(32) void scf_counts_wmma(const float* __restrict__ X,
                                                      const float* __restrict__ w,
                                                      float* __restrict__ counts) {
    const int rowBase = blockIdx.x * 16;
    const int lane    = threadIdx.x;           // 0..31
    const int row     = rowBase + (lane & 15);
    const int khalf   = (lane >> 4) * 2;       // 0 for lanes 0-15, 2 for 16-31
    const float* __restrict__ xrow = X + (size_t)row * N_PANEL + khalf;
    const float* __restrict__ wp   = w + khalf;

#if __has_builtin(__builtin_amdgcn_wmma_f32_16x16x4_f32)
    v8f acc = {};
    for (int k = 0; k + 4 <= N_PANEL; k += 4) {   // N_PANEL % 4 == 0
        v2f a, b;
        a.x = xrow[k];     a.y = xrow[k + 1];
        b.x = wp[k];       b.y = wp[k + 1];
        acc = __builtin_amdgcn_wmma_f32_16x16x4_f32(
            /*neg_a=*/false, a, /*neg_b=*/false, b,
            /*c_mod=*/(short)0, acc, /*reuse_a=*/false, /*reuse_b=*/false);
    }
    // C/D layout: VGPR j: lanes 0-15 -> M=j, lanes 16-31 -> M=8+j.
    if (lane == 0) {
        for (int j = 0; j < 8; ++j) counts[rowBase + j] = acc[j];
    } else if (lane == 16) {
        for (int j = 0; j < 8; ++j) counts[rowBase + 8 + j] = acc[j];
    }
#else
    // Fallback: per-lane partial dot (lanes r and r+16 split K mod 4), shuffle-combine.
    float s = 0.0f;
    for (int k = 0; k + 4 <= N_PANEL; k += 4) {
        s += xrow[k] * wp[k] + xrow[k + 1] * wp[k + 1];
    }
    s += __shfl_xor(s, 16, 32);
    if (lane < 16) counts[rowBase + lane] = s;
#endif
}

// ---------------------------------------------------------------------------
// Kernel 3: lower median of positive counts (single 1024-thread workgroup,
// bitonic sort of 4096 floats in LDS; non-positive -> +inf sorts to the end).
// after = sorted[(npos-1)/2]  (matches torch.median lower-median semantics).
// ---------------------------------------------------------------------------
__global__ __launch_bounds__(1024) void scf_median(const float* __restrict__ counts,
                                                   float* __restrict__ after_out) {
    __shared__ float s[N_CELLS];
    __shared__ int npos;
    const int tid = threadIdx.x;
    if (tid == 0) npos = 0;
    __syncthreads();

    int localpos = 0;
    for (int i = tid; i < N_CELLS; i += 1024) {
        float c = counts[i];
        bool p = c > 0.0f;
        localpos += p ? 1 : 0;
        s[i] = p ? c : INFINITY;
    }
    atomicAdd(&npos, localpos);
    __syncthreads();

    for (int k = 2; k <= N_CELLS; k <<= 1) {
        for (int j = k >> 1; j > 0; j >>= 1) {
            for (int i = tid; i < N_CELLS; i += 1024) {
                int ixj = i ^ j;
                if (ixj > i) {
                    float a = s[i], b = s[ixj];
                    bool up = ((i & k) == 0);
                    if (up ? (a > b) : (a < b)) { s[i] = b; s[ixj] = a; }
                }
            }
            __syncthreads();
        }
    }

    if (tid == 0) {
        int kp  = npos;
        int idx = (kp > 0) ? ((kp - 1) >> 1) : 0;
        after_out[0] = s[idx];
    }
}

// ---------------------------------------------------------------------------
// Kernel 4: main transform. One workgroup per row. The entire 72 KB X row is
// staged into LDS (320 KB/WGP on CDNA5 makes this possible), converting the
// 19264 random gathers into LDS reads. Uses the async global->LDS path when
// the toolchain exposes it.
// ---------------------------------------------------------------------------
__global__ __launch_bounds__(256) void scf_main(const float* __restrict__ X,
                                                const int* __restrict__ gidx,
                                                const unsigned char* __restrict__ missing,
                                                const float* __restrict__ counts,
                                                const float* __restrict__ after_p,
                                                float* __restrict__ out) {
    __shared__ __align__(16) float srow[N_PANEL];   // 72000 bytes
    const int n   = blockIdx.x;
    const int tid = threadIdx.x;
    const float* __restrict__ xrow = X + (size_t)n * N_PANEL;

#if __has_builtin(__builtin_amdgcn_global_load_async_to_lds_b128)
    {
        char* gbase = const_cast<char*>((const char*)xrow);
        char* lbase = (char*)srow;
        for (int off = tid * 16; off < N_PANEL * 4; off += 256 * 16) {
            __builtin_amdgcn_global_load_async_to_lds_b128(
                (gbl4_t)(gbase + off),
                (lds4_t)(lbase + off),
                0, 0);
        }
    }
#if __has_builtin(__builtin_amdgcn_s_wait_asynccnt)
    __builtin_amdgcn_s_wait_asynccnt(0);
#else
    asm volatile("s_wait_asynccnt 0" ::: "memory");
#endif
    __syncthreads();
#else
    {
        const float4* __restrict__ xv = (const float4*)xrow;
        float4* __restrict__       sv = (float4*)srow;
        for (int i = tid; i < N_PANEL / 4; i += 256) sv[i] = xv[i];
    }
    __syncthreads();
#endif

    const float c     = counts[n];
    const float c2    = c + ((c == 0.0f) ? 1.0f : 0.0f);
    const float after = after_p[0];
    const float norm  = c2 / after;                  // matches reference ordering
    float* __restrict__ orow = out + (size_t)n * N_OUTC;

    for (int g = tid; g < N_GENES; g += 256) {
        int   j = gidx[g];
        float v = missing[g] ? 0.0f : srow[j];
        orow[g] = log1pf(v / norm);
    }
    if (tid < 2) {
        orow[N_GENES + tid] = log10f(c2);
    }
}

// ---------------------------------------------------------------------------
// Launcher
// ---------------------------------------------------------------------------
extern "C" void kernel_launch(void* const* d_in, const int* in_sizes, int n_in,
                              void* d_out, int out_size, void* d_ws, size_t ws_size,
                              hipStream_t stream) {
    const float*         X       = (const float*)d_in[0];
    const int*           gidx    = (const int*)d_in[1];
    const unsigned char* missing = (const unsigned char*)d_in[2];  // bool array, 1B/elem
    float* out = (float*)d_out;

    float* w      = (float*)d_ws;            // [N_PANEL]
    float* counts = w + N_PANEL;             // [N_CELLS]
    float* after  = counts + N_CELLS;        // [1]

    scf_zero_w   <<<(N_PANEL + 255) / 256, 256, 0, stream>>>(w);
    scf_build_w  <<<(N_GENES + 255) / 256, 256, 0, stream>>>(gidx, missing, w);
    scf_counts_wmma<<<N_CELLS / 16, 32, 0, stream>>>(X, w, counts);
    scf_median   <<<1, 1024, 0, stream>>>(counts, after);
    scf_main     <<<N_CELLS, 256, 0, stream>>>(X, gidx, missing, counts, after, out);
}